// GetCostVolume_71786083385534
// MI455X (gfx1250) — compile-verified
//
#include <hip/hip_runtime.h>

// Problem constants (from reference): B,C,H,W,D = 4,64,64,192,24
constexpr int NB = 4;
constexpr int NC = 64;
constexpr int NH = 64;
constexpr int NW = 192;
constexpr int ND = 24;
constexpr int PLANE = NH * NW;          // 12288 (per-channel plane)
constexpr int NCHW = NB * NC * PLANE;   // one full [B,C,H,W] tensor (floats)
constexpr int NTILE_N = PLANE / 16;     // 768 tiles along pixels
constexpr int NTILE_M = NC / 16;        // 4 tiles along output channels

typedef float v2f __attribute__((ext_vector_type(2)));
typedef float v8f __attribute__((ext_vector_type(8)));

// -------------------------------------------------------------------------
// Kernel 1: fused 1x1 convs (Q1,K1,Q2,K2) as WMMA f32 16x16x4 GEMMs.
//   out[o,p] = sum_i W[o,i] * X[i,p] + bias[o], per batch, X row stride=PLANE
// One wave per 16x16 output tile; K-loop of 16 WMMAs.
// -------------------------------------------------------------------------
__global__ __launch_bounds__(256) void conv1x1_wmma_kernel(
    const float* __restrict__ x, const float* __restrict__ y,
    const float* __restrict__ qw, const float* __restrict__ qb,
    const float* __restrict__ kwm, const float* __restrict__ kbm,
    float* __restrict__ ws)
{
    const int wave = (blockIdx.x * blockDim.x + threadIdx.x) >> 5;
    const int lane = threadIdx.x & 31;
    const int half = lane >> 4;     // 0: lanes 0-15, 1: lanes 16-31
    const int l16  = lane & 15;

    // tile decomposition: wave -> (nt, mt, b, mat)
    const int nt   = wave % NTILE_N;
    int rest       = wave / NTILE_N;
    const int mt   = rest & 3;                // M tile (4)
    const int g    = rest >> 2;               // 0..15
    const int b    = g & 3;
    const int mat  = g >> 2;                  // 0:Q1 1:K1 2:Q2 3:K2

    const float* X  = ((mat == 0) || (mat == 3)) ? x : y;  // Q1,K2 from x
    const float* Wm = (mat & 1) ? kwm : qw;                // K* use kw
    const float* Bv = (mat & 1) ? kbm : qb;
    X += (size_t)b * NC * PLANE;
    float* O = ws + (size_t)mat * NCHW + (size_t)b * NC * PLANE;

    const int m0 = mt * 16;
    const int n0 = nt * 16;

    v8f acc = {};
    #pragma unroll
    for (int k0 = 0; k0 < NC; k0 += 4) {
        const int kA = k0 + half * 2;         // this half-wave's K pair
        // A: weight tile 16x4, lane holds row m0+l16, K = kA, kA+1
        const float* ap = Wm + (m0 + l16) * NC + kA;
        v2f a; a.x = ap[0]; a.y = ap[1];
        // B: input tile 4x16, lane holds col n0+l16, K = kA, kA+1
        const float* bp = X + (size_t)kA * PLANE + n0 + l16;
        v2f bv; bv.x = bp[0]; bv.y = bp[PLANE];
        acc = __builtin_amdgcn_wmma_f32_16x16x4_f32(
            false, a, false, bv, (short)0, acc, false, false);
    }

    // C/D layout: VGPR r -> M = m0 + r + 8*half, N = n0 + l16
    const int mbase = m0 + half * 8;
    const int nOut  = n0 + l16;
    #pragma unroll
    for (int r = 0; r < 8; ++r) {
        const int m = mbase + r;
        O[(size_t)m * PLANE + nOut] = acc[r] + Bv[m];
    }
}

// -------------------------------------------------------------------------
// Kernel 2: cost volume for one (Q,K,disp) triple.
// Block per (b,h) row. ys depends only on h -> only 2 K rows needed.
// C processed in 4 chunks of 16 channels staged in LDS (3 * 16*192 * 4B = 36KB).
// Each thread owns 18 (d,w) outputs; bilinear x-weights cached in registers.
// -------------------------------------------------------------------------
__global__ __launch_bounds__(256) void cost_volume_kernel(
    const float* __restrict__ Q, const float* __restrict__ K,
    const float* __restrict__ disp, float* __restrict__ out)
{
    __shared__ float Qs [16 * NW];
    __shared__ float K0s[16 * NW];
    __shared__ float K1s[16 * NW];

    const int bh = blockIdx.x;
    const int b  = bh / NH;
    const int h  = bh % NH;
    const int tid = threadIdx.x;

    // Row-uniform y interpolation params (align_corners=False unnormalize)
    const float ys  = (float)h * ((float)NH / (float)(NH - 1)) - 0.5f;
    const float y0f = floorf(ys);
    const float ty  = ys - y0f;
    const int y0 = (int)y0f;
    const int y1 = y0 + 1;
    const float wy0 = (1.f - ty) * ((y0 >= 0 && y0 < NH) ? 1.f : 0.f);
    const float wy1 = ty         * ((y1 >= 0 && y1 < NH) ? 1.f : 0.f);
    const int y0c = min(max(y0, 0), NH - 1);
    const int y1c = min(max(y1, 0), NH - 1);

    constexpr int NE = (ND * NW) / 256;   // 18 outputs per thread
    float accv[NE];
    float wx0a[NE], wx1a[NE];
    int   xpk[NE];

    #pragma unroll
    for (int j = 0; j < NE; ++j) {
        const int idx = j * 256 + tid;
        const int d = idx / NW;
        const int w = idx % NW;
        const float xs  = disp[(((size_t)b * ND + d) * NH + h) * NW + w]
                          * ((float)NW / (float)(NW - 1)) - 0.5f;
        const float x0f = floorf(xs);
        const float tx  = xs - x0f;
        const int x0 = (int)x0f;
        const int x1 = x0 + 1;
        wx0a[j] = (1.f - tx) * ((x0 >= 0 && x0 < NW) ? 1.f : 0.f);
        wx1a[j] = tx         * ((x1 >= 0 && x1 < NW) ? 1.f : 0.f);
        const int x0c = min(max(x0, 0), NW - 1);
        const int x1c = min(max(x1, 0), NW - 1);
        xpk[j] = x0c | (x1c << 16);
        accv[j] = 0.f;
    }

    const float* Qb  = Q + (size_t)b * NC * PLANE + h * NW;
    const float* K0b = K + (size_t)b * NC * PLANE + y0c * NW;
    const float* K1b = K + (size_t)b * NC * PLANE + y1c * NW;

    for (int cc = 0; cc < NC; cc += 16) {
        __syncthreads();
        for (int i = tid; i < 16 * NW; i += 256) {
            const int c = i / NW;
            const int w = i % NW;
            const size_t co = (size_t)(cc + c) * PLANE + w;
            Qs[i]  = Qb[co];
            K0s[i] = K0b[co];
            K1s[i] = K1b[co];
        }
        __syncthreads();

        for (int j = 0; j < NE; ++j) {
            const int idx = j * 256 + tid;
            const int w   = idx % NW;
            const int x0c = xpk[j] & 0xffff;
            const int x1c = xpk[j] >> 16;
            const float wx0 = wx0a[j];
            const float wx1 = wx1a[j];
            float a = 0.f;
            #pragma unroll
            for (int c = 0; c < 16; ++c) {
                const float k0v = wx0 * K0s[c * NW + x0c] + wx1 * K0s[c * NW + x1c];
                const float k1v = wx0 * K1s[c * NW + x0c] + wx1 * K1s[c * NW + x1c];
                a = fmaf(Qs[c * NW + w], wy0 * k0v + wy1 * k1v, a);
            }
            accv[j] += a;
        }
    }

    constexpr float invC = 1.f / (float)NC;
    #pragma unroll
    for (int j = 0; j < NE; ++j) {
        const int idx = j * 256 + tid;
        const int d = idx / NW;
        const int w = idx % NW;
        out[(((size_t)b * ND + d) * NH + h) * NW + w] = accv[j] * invC;
    }
}

// -------------------------------------------------------------------------
// Launch: conv1x1 (WMMA) -> workspace {Q1,K1,Q2,K2}, then 2 cost volumes.
// -------------------------------------------------------------------------
extern "C" void kernel_launch(void* const* d_in, const int* in_sizes, int n_in,
                              void* d_out, int out_size, void* d_ws, size_t ws_size,
                              hipStream_t stream) {
    const float* x  = (const float*)d_in[0];
    const float* y  = (const float*)d_in[1];
    const float* d1 = (const float*)d_in[2];
    const float* d2 = (const float*)d_in[3];
    // d_in[4] = ndisp (int scalar), fixed to 24 here
    const float* qw = (const float*)d_in[5];
    const float* qb = (const float*)d_in[6];
    const float* kw = (const float*)d_in[7];
    const float* kb = (const float*)d_in[8];

    float* ws  = (float*)d_ws;
    float* out = (float*)d_out;

    // 16 GEMMs (4 outputs x 4 batches), 4x768 tiles each = 49152 waves
    const int tiles  = NTILE_N * NTILE_M * (NB * 4);
    const int blocks = tiles / 8;   // 8 waves (256 threads) per block
    conv1x1_wmma_kernel<<<blocks, 256, 0, stream>>>(x, y, qw, qb, kw, kb, ws);

    const float* Q1 = ws + 0 * (size_t)NCHW;
    const float* K1 = ws + 1 * (size_t)NCHW;
    const float* Q2 = ws + 2 * (size_t)NCHW;
    const float* K2 = ws + 3 * (size_t)NCHW;

    const size_t volElems = (size_t)NB * ND * NH * NW;
    cost_volume_kernel<<<NB * NH, 256, 0, stream>>>(Q1, K1, d1, out);
    cost_volume_kernel<<<NB * NH, 256, 0, stream>>>(Q2, K2, d2, out + volElems);
}